// RadialTokenizer_65807488910023
// MI455X (gfx1250) — compile-verified
//
#include <hip/hip_runtime.h>
#include <hip/hip_bf16.h>

typedef __attribute__((ext_vector_type(2))) float v2f;
typedef __attribute__((ext_vector_type(8))) float v8f;

#define NRINGS 16
#define NBINS  256

// One workgroup per (img, ch) plane. 256 threads = 8 waves (wave32).
// Phase 1: stream the 256x256 plane (float4 coalesced), build a 16x256 LDS
//          histogram of the quantized values with ds_add atomics.
// Phase 2: per-ring counts + weighted moments via V_WMMA_F32_16X16X4_F32
//          (D = hist(16x256-chunked) x [v, v^2] weights).
// Phase 3: mean/std from moments, median from histogram order statistics.
__global__ __launch_bounds__(256) void radial_tokenizer_kernel(
    const float* __restrict__ in, float* __restrict__ out) {
    __shared__ unsigned int s_hist[NRINGS * NBINS];   // 16 KB
    __shared__ float        s_sum[NRINGS];
    __shared__ float        s_sq[NRINGS];
    __shared__ unsigned int s_cnt[NRINGS];

    const int t   = threadIdx.x;
    const int ch  = blockIdx.x;   // 0..2
    const int img = blockIdx.y;   // 0..255

    for (int i = t; i < NRINGS * NBINS; i += 256) s_hist[i] = 0u;
    if (t < NRINGS) { s_cnt[t] = 0u; s_sum[t] = 0.0f; s_sq[t] = 0.0f; }
    __syncthreads();

    // ---- Phase 1: histogram of the quantized plane ----
    const float4* plane = (const float4*)(in + (((size_t)img * 3 + ch) << 16));
    for (int iter = 0; iter < 64; ++iter) {
        float4 q = plane[iter * 256 + t];            // b128 per lane, coalesced
        int pbase = (iter * 256 + t) * 4;
        float vals[4] = {q.x, q.y, q.z, q.w};
#pragma unroll
        for (int e = 0; e < 4; ++e) {
            int p = pbase + e;
            int y = p >> 8, x = p & 255;
            int dx = x - 128, dy = y - 128;
            int d2 = dx * dx + dy * dy;
            if (d2 > 0 && d2 <= 16384) {             // inside rings, center excluded
                float h  = vals[e] * 0.5f + 0.5f;
                int  val = (int)floorf(h * 255.0f);
                val = val < 0 ? 0 : (val > 255 ? 255 : val);
                // ring r: 64 r^2 < d2 <= 64 (r+1)^2  (sqrt seed + exact int fixup)
                int r = (int)(sqrtf((float)d2) * 0.125f);
                if (r > 0 && d2 <= 64 * r * r) --r;
                if (d2 > 64 * (r + 1) * (r + 1)) ++r;
                if (r > 15) r = 15;
                atomicAdd(&s_hist[(r << 8) + val], 1u);
            }
        }
    }
    __syncthreads();

    // ---- Phase 2a: per-ring element counts (16 partial sums per ring) ----
    {
        int r = t >> 4;
        int cbase = (t & 15) << 4;
        unsigned int part = 0;
#pragma unroll
        for (int j = 0; j < 16; ++j) part += s_hist[(r << 8) + cbase + j];
        atomicAdd(&s_cnt[r], part);
    }

    // ---- Phase 2b: weighted moments via WMMA f32 16x16x4 ----
    // A[m][k] = hist[ring m][bin k] ; B[k][0] = k, B[k][1] = k^2, else 0.
    // Accumulate K=256 in chunks of 4 -> D[m][0] = sum v*h, D[m][1] = sum v^2*h.
    // All waves run the WMMA uniformly (EXEC must be all ones for WMMA).
    {
        const int lane  = t & 31;
        const int mrow  = lane & 15;        // A row m / B column n
        const int khalf = (lane >> 4) << 1; // lanes 0-15: K base 0; 16-31: K base 2
        v8f acc = {};
        for (int k0 = 0; k0 < NBINS; k0 += 4) {
            int ka = k0 + khalf;
            v2f a, b;
            a.x = (float)s_hist[(mrow << 8) + ka];
            a.y = (float)s_hist[(mrow << 8) + ka + 1];
            float k0f = (float)ka, k1f = (float)(ka + 1);
            b.x = (mrow == 0) ? k0f : ((mrow == 1) ? k0f * k0f : 0.0f);
            b.y = (mrow == 0) ? k1f : ((mrow == 1) ? k1f * k1f : 0.0f);
            acc = __builtin_amdgcn_wmma_f32_16x16x4_f32(
                false, a, false, b, (short)0, acc, false, false);
        }
        // C/D layout: VGPR j holds (M=j, N=lane) for lanes 0-15 and (M=j+8) for 16-31.
        // Column N=0 -> sum, N=1 -> sumsq. Extract from wave 0 lanes 0/1/16/17.
        if (t == 0 || t == 16) {
            int mo = (t == 16) ? 8 : 0;
#pragma unroll
            for (int j = 0; j < 8; ++j) s_sum[mo + j] = acc[j];
        }
        if (t == 1 || t == 17) {
            int mo = (t == 17) ? 8 : 0;
#pragma unroll
            for (int j = 0; j < 8; ++j) s_sq[mo + j] = acc[j];
        }
    }
    __syncthreads();

    // ---- Phase 3: finalize mean/std/median, one thread per ring ----
    if (t < NRINGS) {
        int r = t;
        unsigned int n = s_cnt[r];
        float mean = 0.0f, sd = 0.0f, med = 0.0f;
        if (n > 0) {
            float inv = 1.0f / (float)n;
            mean = s_sum[r] * inv;
            float var = s_sq[r] * inv - mean * mean;
            sd = sqrtf(var > 0.0f ? var : 0.0f);
            // jnp.median: avg of order stats at ranks (n-1)/2 and n/2 (0-based)
            unsigned int k1 = (n - 1) >> 1, k2 = n >> 1;
            unsigned int cum = 0;
            int v1 = -1, v2 = -1;
            for (int v = 0; v < NBINS; ++v) {
                cum += s_hist[(r << 8) + v];
                if (v1 < 0 && cum > k1) v1 = v;
                if (v2 < 0 && cum > k2) { v2 = v; break; }
            }
            med = 0.5f * ((float)v1 + (float)v2);
        }
        float* o = out + ((size_t)img * NRINGS + r) * 9;
        o[ch]     = mean;
        o[3 + ch] = sd;
        o[6 + ch] = med;
    }
}

extern "C" void kernel_launch(void* const* d_in, const int* in_sizes, int n_in,
                              void* d_out, int out_size, void* d_ws, size_t ws_size,
                              hipStream_t stream) {
    (void)in_sizes; (void)n_in; (void)d_ws; (void)ws_size; (void)out_size;
    const float* in = (const float*)d_in[0];
    float* out = (float*)d_out;
    dim3 grid(3, 256);   // (ch, img): 768 workgroups, one 256KB plane each
    radial_tokenizer_kernel<<<grid, dim3(256), 0, stream>>>(in, out);
}